// SoftmaxAttentionAr_6640019439658
// MI455X (gfx1250) — compile-verified
//
#include <hip/hip_runtime.h>

#define B_   16
#define N_   8
#define E_   2048
#define H_   16
#define D_   128
#define KV_  4096
#define KVN_ 4104
#define F_   6144

typedef __attribute__((ext_vector_type(2))) float v2f;
typedef __attribute__((ext_vector_type(8))) float v8f;

#if __has_builtin(__builtin_amdgcn_global_load_async_to_lds_b128) && \
    __has_builtin(__builtin_amdgcn_global_store_async_from_lds_b128) && \
    __has_builtin(__builtin_amdgcn_s_wait_asynccnt)
#define HAVE_ASYNC_LDS 1
#else
#define HAVE_ASYNC_LDS 0
#endif

// The async-to-LDS builtins take int4 pointers: global side in AS(1)
// ("__device__"), LDS side in AS(3).  Cast through uintptr_t; for LDS the
// low 32 bits of a generic pointer are the LDS offset (ISA 10.2 apertures).
typedef __attribute__((__vector_size__(16))) int i128v;
typedef __attribute__((address_space(1))) i128v g_i128;
typedef __attribute__((address_space(3))) i128v l_i128;

__device__ __forceinline__ g_i128* gcast(const float* p) {
  return (g_i128*)(uintptr_t)p;
}
__device__ __forceinline__ l_i128* lcast(float* p) {
  return (l_i128*)(unsigned)(uintptr_t)p;
}

__device__ __forceinline__ v8f wmma4(v2f a, v2f b, v8f c) {
  // D = A(16x4, f32) * B(4x16, f32) + C(16x16, f32)
  return __builtin_amdgcn_wmma_f32_16x16x4_f32(false, a, false, b, (short)0, c,
                                               false, false);
}
__device__ __forceinline__ v2f ldg2(const float* p) {
  return *reinterpret_cast<const v2f*>(p);
}

// ---------------------------------------------------------------------------
// 1) QKV projection: qkv[m][f] = sum_e x[m][e] * Wq[f][e]
//    M=128 rows (b*N+n), K=2048, F=6144.  One wave per 16x16 tile.
//    Writes q -> ws[B,H,N,D]; k_new/v_new -> cache rows KV..KV+N-1.
// ---------------------------------------------------------------------------
__global__ __launch_bounds__(256) void qkv_kernel(
    const float* __restrict__ x, const float* __restrict__ Wq,
    float* __restrict__ qws, float* __restrict__ kout, float* __restrict__ vout) {
  const int lane = threadIdx.x & 31;
  const int wv   = threadIdx.x >> 5;
  const int f0   = blockIdx.x * 16;
  const int m0   = wv * 16;
  const int lo   = lane & 15;
  const int hi   = lane >> 4;

  const float* xrow = x  + (size_t)(m0 + lo) * E_ + 2 * hi;
  const float* wrow = Wq + (size_t)(f0 + lo) * E_ + 2 * hi;

  v8f c = {};
#pragma unroll 8
  for (int k = 0; k < E_; k += 4) {
    v2f a  = ldg2(xrow + k);
    v2f bb = ldg2(wrow + k);
    c = wmma4(a, bb, c);
  }

  const int f = f0 + lo;
#pragma unroll
  for (int i = 0; i < 8; i++) {
    int m = m0 + i + 8 * hi;         // C layout: vgpr i -> row i (lo half) / i+8
    int b = m >> 3, n = m & 7;
    float v = c[i];
    if (f < E_) {
      int h = f >> 7, d = f & 127;
      qws[(((size_t)b * H_ + h) * N_ + n) * D_ + d] = v;
    } else if (f < 2 * E_) {
      int fe = f - E_;
      int h = fe >> 7, d = fe & 127;
      kout[(((size_t)b * H_ + h) * KVN_ + KV_ + n) * D_ + d] = v;
    } else {
      int fe = f - 2 * E_;
      int h = fe >> 7, d = fe & 127;
      vout[(((size_t)b * H_ + h) * KVN_ + KV_ + n) * D_ + d] = v;
    }
  }
}

// ---------------------------------------------------------------------------
// 2) Fused KV-cache-copy + flash attention.
//    One block (8 waves) per (b,h).  Each wave owns key tiles t, t+8, ...
//    Per tile: async-DMA K/V (16x128 f32) from past_k/past_v into LDS,
//    async-store the same LDS tile out to the returned cache (overlapped with
//    compute), and feed WMMA operands from LDS.  Past K/V touches HBM once.
// ---------------------------------------------------------------------------
__global__ __launch_bounds__(256) void attn_fused_kernel(
    const float* __restrict__ qws,
    const float* __restrict__ pastk, const float* __restrict__ pastv,
    float* __restrict__ kc, float* __restrict__ vc,
    float* __restrict__ attnout) {

  // 38976 floats = 152.25 KB (<= 320 KB/WG on CDNA5); merge buffers overlay
  // the staging region after the main loop.
  __shared__ __align__(16) float smem[38976];
  float* Qs  = smem;                  // [16][132]   (pad 132: banks 4r)
  float* Psm = smem + 2112;           // [8][16][20] per-wave P bounce
  float* Kst = smem + 4672;           // [8][16][132]
  float* Vst = Kst + 8 * 16 * 132;    // [8][16][136] (pad 136: row-pair b32 ok)
  float* Osh = Kst;                   // overlay: [8][8][128]
  float* msh = Kst + 8192;            // overlay: [8][8]
  float* lsh = msh + 64;              // overlay: [8][8]

  const int tid  = threadIdx.x;
  const int lane = tid & 31;
  const int w    = tid >> 5;
  const int bh   = blockIdx.x;        // b*H + h
  const int b    = bh >> 4;
  const int h    = bh & 15;

  float*       kbh  = kc    + (size_t)bh * KVN_ * D_;
  float*       vbh  = vc    + (size_t)bh * KVN_ * D_;
  const float* pkbh = pastk + (size_t)bh * KV_  * D_;
  const float* pvbh = pastv + (size_t)bh * KV_  * D_;
  const float* qbh  = qws   + (size_t)bh * N_   * D_;

  for (int t = tid; t < 16 * D_; t += 256) {
    int r = t >> 7, d = t & 127;
    Qs[r * 132 + d] = (r < N_) ? qbh[r * D_ + d] : 0.0f;  // pad q rows with 0
  }
  __syncthreads();

  const int lo    = lane & 15;
  const int hi    = lane >> 4;
  const int koff0 = 2 * hi;
  const int lane4 = lane * 4;

  float* KstW = Kst + w * (16 * 132);
  float* VstW = Vst + w * (16 * 136);

  float mrow[8], lrow[8];
  v8f acc[8] = {};
#pragma unroll
  for (int i = 0; i < 8; i++) { mrow[i] = -3.0e38f; lrow[i] = 0.0f; }

  const float scale = 0.088388347648318447f;  // 1/sqrt(128)
  const int NT = (KVN_ + 15) / 16;            // 257 key tiles

  for (int t = w; t < NT; t += 8) {
    const int  kt0   = t * 16;
    const bool full  = (t < KV_ / 16);        // tiles 0..255: past rows
    const int  nrows = full ? 16 : (KVN_ - kt0);
    const float* ksrc = full ? (pkbh + (size_t)kt0 * D_) : (kbh + (size_t)kt0 * D_);
    const float* vsrc = full ? (pvbh + (size_t)kt0 * D_) : (vbh + (size_t)kt0 * D_);

    // ---- stage K/V tile in LDS; copy past rows out to the returned cache
#if HAVE_ASYNC_LDS
    __builtin_amdgcn_s_wait_asynccnt(0);  // staging free (prev copy-out done)
    for (int r = 0; r < nrows; ++r)
      __builtin_amdgcn_global_load_async_to_lds_b128(
          gcast(ksrc + r * D_ + lane4), lcast(KstW + r * 132 + lane4), 0, 0);
    for (int r = 0; r < nrows; ++r)
      __builtin_amdgcn_global_load_async_to_lds_b128(
          gcast(vsrc + r * D_ + lane4), lcast(VstW + r * 136 + lane4), 0, 0);
    __builtin_amdgcn_s_wait_asynccnt(0);  // tile resident in LDS
    if (full) {                            // overlaps with compute below
      for (int r = 0; r < 16; ++r)
        __builtin_amdgcn_global_store_async_from_lds_b128(
            gcast(kbh + (size_t)(kt0 + r) * D_ + lane4),
            lcast(KstW + r * 132 + lane4), 0, 0);
      for (int r = 0; r < 16; ++r)
        __builtin_amdgcn_global_store_async_from_lds_b128(
            gcast(vbh + (size_t)(kt0 + r) * D_ + lane4),
            lcast(VstW + r * 136 + lane4), 0, 0);
    }
#else
    for (int r = 0; r < nrows; ++r) {
      float4 kk = *(const float4*)(ksrc + r * D_ + lane4);
      float4 vv = *(const float4*)(vsrc + r * D_ + lane4);
      *(float4*)&KstW[r * 132 + lane4] = kk;
      *(float4*)&VstW[r * 136 + lane4] = vv;
      if (full) {
        *(float4*)(kbh + (size_t)(kt0 + r) * D_ + lane4) = kk;
        *(float4*)(vbh + (size_t)(kt0 + r) * D_ + lane4) = vv;
      }
    }
#endif

    // ---- S = Q * K^T over d=0..127 in steps of 4 (32 WMMAs, LDS operands)
    v8f s = {};
#pragma unroll 4
    for (int k = 0; k < D_; k += 4) {
      v2f a  = *reinterpret_cast<const v2f*>(&Qs[lo * 132 + k + koff0]);
      v2f bb = *reinterpret_cast<const v2f*>(&KstW[lo * 132 + k + koff0]);
      s = wmma4(a, bb, s);
    }

    // mask invalid key columns (stale staging rows on the last tile)
    const float colbias = (kt0 + lo < KVN_) ? 0.0f : -3.0e38f;

    // ---- online softmax; C layout: vgpr i == row i (lanes 0-15) / i+8 (16-31)
#pragma unroll
    for (int i = 0; i < 8; i++) {
      float si = s[i] * scale + colbias;
      float tm = si;
#pragma unroll
      for (int off = 1; off < 16; off <<= 1)
        tm = fmaxf(tm, __shfl_xor(tm, off, 32));
      float mn = fmaxf(mrow[i], tm);
      float al = __expf(mrow[i] - mn);
      float p  = __expf(si - mn);
      float ps = p;
#pragma unroll
      for (int off = 1; off < 16; off <<= 1)
        ps += __shfl_xor(ps, off, 32);
      lrow[i] = lrow[i] * al + ps;
      mrow[i] = mn;
#pragma unroll
      for (int j = 0; j < 8; j++) acc[j][i] *= al;
      Psm[(w * 16 + i + 8 * hi) * 20 + lo] = p;  // C-layout -> A-layout bounce
    }

    // ---- O += P * V  (16x16 * 16x128, 32 WMMAs, LDS operands)
#pragma unroll
    for (int c = 0; c < 4; c++) {
      v2f a = *reinterpret_cast<const v2f*>(&Psm[(w * 16 + lo) * 20 + 4 * c + koff0]);
      const int r0 = 4 * c + koff0;
#pragma unroll
      for (int j = 0; j < 8; j++) {
        const int col = j * 16 + lo;
        v2f bb;
        bb.x = VstW[r0 * 136 + col];
        bb.y = VstW[(r0 + 1) * 136 + col];
        acc[j] = wmma4(a, bb, acc[j]);
      }
    }
  }

#if HAVE_ASYNC_LDS
  __builtin_amdgcn_s_wait_asynccnt(0);  // copy-out done before staging reuse
#endif
  __syncthreads();                       // all waves done with staging region

  // ---- stash per-wave partials (valid rows 0..7 live in lanes 0-15)
  if (lane < 16) {
#pragma unroll
    for (int j = 0; j < 8; j++)
#pragma unroll
      for (int i = 0; i < 8; i++)
        Osh[(w * 8 + i) * 128 + j * 16 + lo] = acc[j][i];
  }
  if (lane == 0) {
#pragma unroll
    for (int i = 0; i < 8; i++) { msh[w * 8 + i] = mrow[i]; lsh[w * 8 + i] = lrow[i]; }
  }
  __syncthreads();

  // ---- cross-wave log-sum-exp merge; write [B,N,E]
  for (int e = tid; e < N_ * D_; e += 256) {
    int r = e >> 7, d = e & 127;
    float ms = -3.0e38f;
#pragma unroll
    for (int ww = 0; ww < 8; ww++) ms = fmaxf(ms, msh[ww * 8 + r]);
    float num = 0.0f, den = 0.0f;
#pragma unroll
    for (int ww = 0; ww < 8; ww++) {
      float f = __expf(msh[ww * 8 + r] - ms);
      num += f * Osh[(ww * 8 + r) * 128 + d];
      den += f * lsh[ww * 8 + r];
    }
    attnout[((size_t)b * N_ + r) * E_ + h * D_ + d] = num / den;
  }
}

// ---------------------------------------------------------------------------
// 3) Output projection: out[m][f] = sum_e attn[m][e] * Wo[f][e]
// ---------------------------------------------------------------------------
__global__ __launch_bounds__(256) void oproj_kernel(
    const float* __restrict__ a_in, const float* __restrict__ Wo,
    float* __restrict__ out) {
  const int lane = threadIdx.x & 31;
  const int wv   = threadIdx.x >> 5;
  const int f0   = blockIdx.x * 16;
  const int m0   = wv * 16;
  const int lo   = lane & 15;
  const int hi   = lane >> 4;

  const float* arow = a_in + (size_t)(m0 + lo) * E_ + 2 * hi;
  const float* wrow = Wo   + (size_t)(f0 + lo) * E_ + 2 * hi;

  v8f c = {};
#pragma unroll 8
  for (int k = 0; k < E_; k += 4) {
    v2f a  = ldg2(arow + k);
    v2f bb = ldg2(wrow + k);
    c = wmma4(a, bb, c);
  }
#pragma unroll
  for (int i = 0; i < 8; i++) {
    int m = m0 + i + 8 * hi;
    out[(size_t)m * E_ + f0 + lo] = c[i];
  }
}

// ---------------------------------------------------------------------------
extern "C" void kernel_launch(void* const* d_in, const int* in_sizes, int n_in,
                              void* d_out, int out_size, void* d_ws, size_t ws_size,
                              hipStream_t stream) {
  const float* x  = (const float*)d_in[0];
  const float* pk = (const float*)d_in[1];
  const float* pv = (const float*)d_in[2];
  const float* Wq = (const float*)d_in[3];
  const float* Wo = (const float*)d_in[4];

  float* out  = (float*)d_out;                          // [B,N,E]
  float* kout = out + (size_t)B_ * N_ * E_;             // [B,H,KVN,D]
  float* vout = kout + (size_t)B_ * H_ * KVN_ * D_;     // [B,H,KVN,D]

  float* qws    = (float*)d_ws;                         // [B,H,N,D]
  float* attnws = qws + (size_t)B_ * N_ * E_;           // [B,N,E]

  // 1) fused QKV projection (WMMA f32); also fills cache rows KV..KV+N-1
  qkv_kernel<<<dim3(F_ / 16), dim3(256), 0, stream>>>(x, Wq, qws, kout, vout);
  // 2) fused cache-copy + flash attention (async LDS DMA + WMMA f32)
  attn_fused_kernel<<<dim3(B_ * H_), dim3(256), 0, stream>>>(
      qws, pk, pv, kout, vout, attnws);
  // 3) output projection (WMMA f32)
  oproj_kernel<<<dim3(E_ / 16), dim3(256), 0, stream>>>(attnws, Wo, out);
}